// LocalSTD_22660247453708
// MI455X (gfx1250) — compile-verified
//
#include <hip/hip_runtime.h>
#include <math.h>

typedef __attribute__((ext_vector_type(2))) float v2f;
typedef __attribute__((ext_vector_type(8))) float v8f;

#if defined(__HIP_DEVICE_COMPILE__) && !__has_builtin(__builtin_amdgcn_wmma_f32_16x16x4_f32)
#error "device pass: missing __builtin_amdgcn_wmma_f32_16x16x4_f32"
#endif

#define IMG      256
#define RAD      5
// WG output tile: 64 wide x 32 tall, 8 waves of one 16x16 WMMA tile each
#define IN_ROWS  48      // tr(max 16) + 32 input rows per chunk pair
#define IN_COLS  76      // 64 + 5 left halo + 7 (5 halo + 2 K-pad)
#define INS      79      // LDS stride: 15 = 79 mod 64 -> conflict-free 16-row gather
#define HST      17      // H buffer stride (conflict-free)

// 1D Gaussian, sigma=1, normalized so outer(g,g) == normalized 2D kernel
// h[k] = exp(-k^2/2), S = sum(h) = 2.5066282575371262
#define GS 2.5066282575371262
__constant__ float c_g[11] = {
    (float)(3.726653172078671e-06 / GS), (float)(0.00033546262790251185 / GS),
    (float)(0.011108996538242306 / GS),  (float)(0.1353352832366127 / GS),
    (float)(0.6065306597126334 / GS),    (float)(1.0 / GS),
    (float)(0.6065306597126334 / GS),    (float)(0.1353352832366127 / GS),
    (float)(0.011108996538242306 / GS),  (float)(0.00033546262790251185 / GS),
    (float)(3.726653172078671e-06 / GS)
};

__global__ __launch_bounds__(256) void localstd_wmma_kernel(
    const float* __restrict__ x, float* __restrict__ out)
{
    __shared__ float sIn[IN_ROWS * INS];            // 15168 B
    __shared__ float sH[8][2][32 * HST];            // 34816 B  (per-wave H / H2)

    const int plane = blockIdx.z;
    const int r0 = blockIdx.y * 32;
    const int c0 = blockIdx.x * 64;
    const long pbase = (long)plane * (IMG * IMG);
    const float* __restrict__ xin = x + pbase;

    // ---- cooperative zero-padded halo load: rows r0-5..r0+42, cols c0-5..c0+70
    for (int i = threadIdx.x; i < IN_ROWS * IN_COLS; i += 256) {
        int lr = i / IN_COLS;
        int lc = i - lr * IN_COLS;
        int gr = r0 - RAD + lr;
        int gc = c0 - RAD + lc;
        float v = 0.0f;
        if ((unsigned)gr < (unsigned)IMG && (unsigned)gc < (unsigned)IMG)
            v = xin[gr * IMG + gc];
        sIn[lr * INS + lc] = v;
    }
    __syncthreads();

    const int wave = threadIdx.x >> 5;
    const int lane = threadIdx.x & 31;
    const int m    = lane & 15;          // A-row / B-col index for this lane
    const int kh   = (lane >> 4) * 2;    // K sub-index base within a K=4 step
    const int tr   = (wave >> 2) * 16;   // 0 or 16
    const int tc   = (wave & 3) * 16;    // 0,16,32,48

    // Banded Toeplitz operand: W[t] holds g[k-m] for k = 4t+kh+{0,1}.
    // Identical values serve as horizontal-B (B[k,n]=g[k-n]) and vertical-A
    // (A[m,k]=g[k-m]) under the 32-bit WMMA lane layouts.
    v2f W[7];
#pragma unroll
    for (int t = 0; t < 7; ++t) {
        int d0 = 4 * t + kh - m;
        int d1 = d0 + 1;
        float w0 = ((unsigned)d0 <= 10u) ? c_g[d0] : 0.0f;
        float w1 = ((unsigned)d1 <= 10u) ? c_g[d1] : 0.0f;
        v2f w; w.x = w0; w.y = w1;
        W[t] = w;
    }

    float* hX  = &sH[wave][0][0];
    float* hX2 = &sH[wave][1][0];

    // ---- horizontal pass: H rows k=0..31 <=> image rows r0+tr-5+k
    //      chunk c computes H rows 16c..16c+15 from input LDS rows tr+16c+m
#pragma unroll
    for (int c = 0; c < 2; ++c) {
        const float* rp = &sIn[(tr + 16 * c + m) * INS + tc + kh];
        v8f aH  = {};
        v8f aH2 = {};
#pragma unroll
        for (int t = 0; t < 7; ++t) {
            float a0 = rp[4 * t];
            float a1 = rp[4 * t + 1];
            v2f a;  a.x = a0;       a.y = a1;
            v2f a2; a2.x = a0 * a0; a2.y = a1 * a1;
            aH  = __builtin_amdgcn_wmma_f32_16x16x4_f32(false, a,  false, W[t],
                                                        (short)0, aH,  false, false);
            aH2 = __builtin_amdgcn_wmma_f32_16x16x4_f32(false, a2, false, W[t],
                                                        (short)0, aH2, false, false);
        }
        // D layout: VGPR v -> row v + 8*(lane>>4); col = lane&15
        int rb = 16 * c + (lane >> 4) * 8;
#pragma unroll
        for (int v = 0; v < 8; ++v) {
            hX [(rb + v) * HST + m] = aH[v];
            hX2[(rb + v) * HST + m] = aH2[v];
        }
    }
    // same-wave DS ops are in order; compiler inserts s_wait_dscnt for the RAW

    // ---- vertical pass: Out[m,n] = sum_k g[k-m] * H[k][n], K = 28 (k=26,27 zero-coeff)
    v8f aM = {};
    v8f aE = {};
#pragma unroll
    for (int t = 0; t < 7; ++t) {
        int k = 4 * t + kh;
        v2f bx;  bx.x  = hX [k * HST + m];  bx.y  = hX [(k + 1) * HST + m];
        v2f bx2; bx2.x = hX2[k * HST + m];  bx2.y = hX2[(k + 1) * HST + m];
        aM = __builtin_amdgcn_wmma_f32_16x16x4_f32(false, W[t], false, bx,
                                                   (short)0, aM, false, false);
        aE = __builtin_amdgcn_wmma_f32_16x16x4_f32(false, W[t], false, bx2,
                                                   (short)0, aE, false, false);
    }

    // ---- epilogue: var = E[x^2] - mean^2 + eps ; out = sqrt(var)
    float* op = out + pbase + (long)(r0 + tr + (lane >> 4) * 8) * IMG + (c0 + tc + m);
#pragma unroll
    for (int v = 0; v < 8; ++v) {
        float mean = aM[v];
        float ex2  = aE[v];
        float var  = ex2 - mean * mean + 1.0e-6f;
        op[v * IMG] = sqrtf(var);
    }
}

extern "C" void kernel_launch(void* const* d_in, const int* in_sizes, int n_in,
                              void* d_out, int out_size, void* d_ws, size_t ws_size,
                              hipStream_t stream)
{
    const float* xin = (const float*)d_in[0];
    float* out = (float*)d_out;
    const int nplanes = in_sizes[0] >> 16;          // elements / (256*256) = B*C
    dim3 grid(IMG / 64, IMG / 32, nplanes);
    dim3 block(256);
    localstd_wmma_kernel<<<grid, block, 0, stream>>>(xin, out);
}